// InstanceConv_27187142984268
// MI455X (gfx1250) — compile-verified
//
#include <hip/hip_runtime.h>

typedef __attribute__((ext_vector_type(2))) float v2f;
typedef __attribute__((ext_vector_type(8))) float v8f;

#define B_     8
#define C_     64
#define H_     128
#define W_     128
#define OC_    64
#define KR     576        // C_ * 9  (reduction dim)
#define KSTEPS 144        // KR / 4
#define TPX    64         // pixels per workgroup tile (half an image row)
#define NPT    4          // 16-px sub-tiles per wave (A-fragment reuse factor)
#define XS     72         // padded LDS row stride (dwords): 2*72 mod 64 = 16
                          //  -> lanes 16..31 hit banks +16 vs lanes 0..15: conflict-free

// One workgroup = 128 threads = 4 waves; tile = 64 consecutive pixels in one row.
// Wave w owns OC block w (16 OCs) and sweeps all 4 pixel sub-tiles, so each A
// fragment (weights) is loaded once per K-step and reused by 4 WMMAs. This cuts
// L2 weight traffic 4x (288 MB total ~ the 72 MB HBM floor; inp fits in L2).
__global__ __launch_bounds__(128)
void instconv_wmma_f32(const float* __restrict__ inp,
                       const float* __restrict__ mask,
                       const float* __restrict__ conv,
                       const float* __restrict__ bias,
                       float* __restrict__ out,
                       float* __restrict__ mask_out)
{
    __shared__ float Xlds[KR * XS];      // 162 KB masked-patch matrix (padded rows)
    __shared__ float meq[9 * TPX];       // mask-equality per tap/pixel
    __shared__ float norm[TPX];          // 9 / count per pixel

    const int tid  = (int)threadIdx.x;
    const int lane = tid & 31;
    const int wv   = tid >> 5;

    // grid = B * H * (W/64):  blk -> (b, y, x-half)
    const int blk = (int)blockIdx.x;
    const int xt  = blk & 1;
    const int y   = (blk >> 1) & 127;
    const int b   = blk >> 8;
    const int x0  = xt * TPX;

    const float* maskb = mask + (size_t)b * H_ * W_;

    // ---- Phase 1: per-pixel mask equality + normalization + mask_out ----
    if (tid < TPX) {
        const int x  = x0 + tid;
        const float mc = maskb[y * W_ + x];   // center tap always in-bounds
        float cnt = 0.f;
#pragma unroll
        for (int t = 0; t < 9; ++t) {
            const int dy = t / 3 - 1, dx = t % 3 - 1;
            const int yy = y + dy, xx = x + dx;
            // OOB taps see fill = -1.0, never equal to mc (mc in {0..3})
            const float mv = (yy >= 0 && yy < H_ && xx >= 0 && xx < W_)
                                 ? maskb[yy * W_ + xx] : -1.0f;
            const float eq = (mv == mc) ? 1.0f : 0.0f;
            meq[t * TPX + tid] = eq;
            cnt += eq;
        }
        norm[tid] = 9.0f / fmaxf(cnt, 1e-8f);
        mask_out[((size_t)b * H_ + y) * W_ + x] = mc;
    }
    __syncthreads();

    // ---- Phase 2: build X[r][px], r = c*9 + t (masked input patches) ----
    // consecutive threads -> consecutive px -> coalesced global loads and
    // conflict-free LDS stores.
    const float* inpb = inp + (size_t)b * C_ * H_ * W_;
    for (int e = tid; e < KR * TPX; e += 128) {
        const int px = e & (TPX - 1);
        const int r  = e >> 6;               // TPX == 64
        const int c  = r / 9;
        const int t  = r - c * 9;
        const int dy = t / 3 - 1, dx = t % 3 - 1;
        const int yy = y + dy, xx = x0 + px + dx;
        float v = 0.f;
        if (yy >= 0 && yy < H_ && xx >= 0 && xx < W_)
            v = inpb[(c * H_ + yy) * W_ + xx] * meq[t * TPX + px];
        Xlds[r * XS + px] = v;
    }
    __syncthreads();

    // ---- Phase 3: 4x (16oc x 16px) tiles per wave via V_WMMA_F32_16X16X4_F32 ----
    // A (16x4 f32): lane<16 -> row M=lane, (K0,K1); lane>=16 -> (K2,K3)
    // B (4x16 f32): VGPR0 = rows K0 (lanes 0-15) / K2 (lanes 16-31), VGPR1 = K1/K3
    const int ocb   = wv * 16;
    const int mrow  = lane & 15;
    const int khalf = (lane < 16) ? 0 : 2;

    const float* wrow = conv + (size_t)(ocb + mrow) * KR + khalf; // 8B-aligned
    const float* xb   = Xlds + mrow;                              // + pt*16 per tile

    v8f acc0 = {}, acc1 = {}, acc2 = {}, acc3 = {};
#pragma unroll 2
    for (int ks = 0; ks < KSTEPS; ++ks) {
        const int kk = ks * 4;
        const v2f a = *(const v2f*)(wrow + kk);      // one global_load_b64, reused 4x
        const int r0 = (kk + khalf) * XS;
        const int r1 = r0 + XS;
        v2f bb;
        bb.x = xb[r0 +  0]; bb.y = xb[r1 +  0];
        acc0 = __builtin_amdgcn_wmma_f32_16x16x4_f32(false, a, false, bb, (short)0, acc0, false, false);
        bb.x = xb[r0 + 16]; bb.y = xb[r1 + 16];
        acc1 = __builtin_amdgcn_wmma_f32_16x16x4_f32(false, a, false, bb, (short)0, acc1, false, false);
        bb.x = xb[r0 + 32]; bb.y = xb[r1 + 32];
        acc2 = __builtin_amdgcn_wmma_f32_16x16x4_f32(false, a, false, bb, (short)0, acc2, false, false);
        bb.x = xb[r0 + 48]; bb.y = xb[r1 + 48];
        acc3 = __builtin_amdgcn_wmma_f32_16x16x4_f32(false, a, false, bb, (short)0, acc3, false, false);
    }

    // ---- Epilogue: scale by 9/count, add bias, coalesced stores ----
    // C/D layout: VGPR j -> M=j (lanes 0-15), M=j+8 (lanes 16-31); N = lane&15
    v8f accs[NPT] = {acc0, acc1, acc2, acc3};
#pragma unroll
    for (int pt = 0; pt < NPT; ++pt) {
        const int   px  = pt * 16 + mrow;
        const float nrm = norm[px];
        const int   x   = x0 + px;
#pragma unroll
        for (int j = 0; j < 8; ++j) {
            const int oc  = ocb + j + ((lane < 16) ? 0 : 8);
            const float v = accs[pt][j] * nrm + bias[oc];
            out[(((size_t)b * OC_ + oc) * H_ + y) * W_ + x] = v;
        }
    }
}

extern "C" void kernel_launch(void* const* d_in, const int* in_sizes, int n_in,
                              void* d_out, int out_size, void* d_ws, size_t ws_size,
                              hipStream_t stream) {
    const float* inp  = (const float*)d_in[0];
    const float* mask = (const float*)d_in[1];
    const float* conv = (const float*)d_in[2];
    const float* bias = (const float*)d_in[3];

    float* out      = (float*)d_out;
    float* mask_out = out + (size_t)B_ * OC_ * H_ * W_;   // tuple order: (out, mask_out)

    const dim3 grid(B_ * H_ * (W_ / TPX));   // 8 * 128 * 2 = 2048 tiles
    instconv_wmma_f32<<<grid, 128, 0, stream>>>(inp, mask, conv, bias, out, mask_out);
}